// FraudDetectionHybrid_65481071401543
// MI455X (gfx1250) — compile-verified
//
#include <hip/hip_runtime.h>
#include <random>

// ---------------------------------------------------------------------------
// 4-qubit statevector circuit, B=131072.
// Strategy: the 57 non-batched gates collapse into one fixed 16x16 complex
// unitary U (built once on device). Per batch element the encoder produces a
// product state enc (16 complex), then Psi = U * Enc is a complex GEMM
// (16 x 16 x B) executed with V_WMMA_F32_16X16X4_F32 in 16-batch tiles.
// ---------------------------------------------------------------------------

typedef __attribute__((ext_vector_type(2))) float v2f;
typedef __attribute__((ext_vector_type(8))) float v8f;

__device__ __forceinline__ float2 cmul(float2 a, float2 b) {
    return make_float2(a.x * b.x - a.y * b.y, a.x * b.y + a.y * b.x);
}
__device__ __forceinline__ float2 cadd(float2 a, float2 b) {
    return make_float2(a.x + b.x, a.y + b.y);
}

// D = A(16x4 f32) * B(4x16 f32) + C(16x16 f32), wave32 WMMA.
__device__ __forceinline__ v8f wmma4(v2f a, v2f b, v8f c) {
    return __builtin_amdgcn_wmma_f32_16x16x4_f32(
        /*neg_a=*/false, a, /*neg_b=*/false, b,
        /*c_mod=*/(short)0, c, /*reuse_a=*/false, /*reuse_b=*/false);
}

// --------------------------- gate list ------------------------------------
// kind: 0=RX 1=RY 2=RZ 3=CNOT 4=CRX 5=H 6=SX
// encoding: kind[2:0] | w0[4:3] | w1[6:5] | pidx[12:7]
struct GateOps {
    unsigned n;
    unsigned op[64];
};

// --------------------------- kernel 1: build U -----------------------------
__global__ void build_u_kernel(GateOps ops,
                               const float* __restrict__ rp,
                               const float* __restrict__ rx0,
                               const float* __restrict__ ry0,
                               const float* __restrict__ rz0,
                               const float* __restrict__ crx0,
                               float* __restrict__ Ur, float* __restrict__ Ui) {
    int col = threadIdx.x;
    if (col >= 16) return;
    float2 s[16];
    for (int i = 0; i < 16; ++i) s[i] = make_float2(i == col ? 1.0f : 0.0f, 0.0f);

    for (unsigned g = 0; g < ops.n; ++g) {
        unsigned o = ops.op[g];
        int kind = o & 7;
        int w0 = (o >> 3) & 3;
        int w1 = (o >> 5) & 3;
        int p  = (o >> 7) & 63;
        float th = 0.0f;
        if (p < 50)       th = rp[p];
        else if (p == 50) th = rx0[0];
        else if (p == 51) th = ry0[0];
        else if (p == 52) th = rz0[0];
        else if (p == 53) th = crx0[0];
        float sn, c;
        __sincosf(0.5f * th, &sn, &c);

        float2 m00, m01, m10, m11;
        bool oneq = true;
        switch (kind) {
            case 0: m00 = make_float2(c, 0);   m01 = make_float2(0, -sn);
                    m10 = make_float2(0, -sn); m11 = make_float2(c, 0); break;      // RX
            case 1: m00 = make_float2(c, 0);   m01 = make_float2(-sn, 0);
                    m10 = make_float2(sn, 0);  m11 = make_float2(c, 0); break;      // RY
            case 2: m00 = make_float2(c, -sn); m01 = make_float2(0, 0);
                    m10 = make_float2(0, 0);   m11 = make_float2(c, sn); break;     // RZ
            case 5: { float r = 0.70710678118654752f;
                    m00 = make_float2(r, 0);  m01 = make_float2(r, 0);
                    m10 = make_float2(r, 0);  m11 = make_float2(-r, 0); } break;    // H
            case 6: m00 = make_float2(0.5f, 0.5f);  m01 = make_float2(0.5f, -0.5f);
                    m10 = make_float2(0.5f, -0.5f); m11 = make_float2(0.5f, 0.5f); break; // SX
            default: oneq = false; break;
        }
        if (oneq) {
            int str = 8 >> w0;  // wire 0 = MSB
            for (int i = 0; i < 16; ++i) {
                if (!(i & str)) {
                    float2 a = s[i], b = s[i + str];
                    s[i]       = cadd(cmul(m00, a), cmul(m01, b));
                    s[i + str] = cadd(cmul(m10, a), cmul(m11, b));
                }
            }
        } else if (kind == 3) {  // CNOT(control=w0, target=w1)
            int cs = 8 >> w0, ts = 8 >> w1;
            for (int i = 0; i < 16; ++i) {
                if ((i & cs) && !(i & ts)) {
                    float2 t = s[i]; s[i] = s[i + ts]; s[i + ts] = t;
                }
            }
        } else {                 // CRX(control=w0, target=w1)
            int cs = 8 >> w0, ts = 8 >> w1;
            for (int i = 0; i < 16; ++i) {
                if ((i & cs) && !(i & ts)) {
                    float2 a = s[i], b = s[i + ts];
                    // [c, -i*sn; -i*sn, c]
                    s[i]      = make_float2(c * a.x + sn * b.y, c * a.y - sn * b.x);
                    s[i + ts] = make_float2(c * b.x + sn * a.y, c * b.y - sn * a.x);
                }
            }
        }
    }
    for (int r = 0; r < 16; ++r) {
        Ur[r * 16 + col] = s[r].x;
        Ui[r * 16 + col] = s[r].y;
    }
}

// --------------------------- kernel 2: main batch --------------------------
// Encoder spinor per wire: RY(d) RX(c) RZ(b) RY(a) |0>
__device__ __forceinline__ void spinor(float a, float b, float c, float d,
                                       float2& o0, float2& o1) {
    float sa, ca, sb, cb, sc, cc, sd, cd;
    __sincosf(0.5f * a, &sa, &ca);
    __sincosf(0.5f * b, &sb, &cb);
    __sincosf(0.5f * c, &sc, &cc);
    __sincosf(0.5f * d, &sd, &cd);
    float2 u = make_float2(ca * cb, -ca * sb);  // e^{-ib/2} * ca
    float2 w = make_float2(sa * cb,  sa * sb);  // e^{+ib/2} * sa
    // RX(c): u' = cc*u - i*sc*w ; w' = -i*sc*u + cc*w
    float2 u2 = make_float2(cc * u.x + sc * w.y, cc * u.y - sc * w.x);
    float2 w2 = make_float2(sc * u.y + cc * w.x, -sc * u.x + cc * w.y);
    // RY(d)
    o0 = make_float2(cd * u2.x - sd * w2.x, cd * u2.y - sd * w2.y);
    o1 = make_float2(sd * u2.x + cd * w2.x, sd * u2.y + cd * w2.y);
}

__global__ void __launch_bounds__(256)
qsim_main_kernel(const float* __restrict__ feat,
                 const float* __restrict__ Ur, const float* __restrict__ Ui,
                 float* __restrict__ ezbuf, float* __restrict__ partial,
                 int Bsz, int nwaves) {
    const int lane = threadIdx.x & 31;
    const int m = lane & 15;        // A row / batch-in-tile
    const int h = lane >> 4;        // half-wave (K split)
    const int wave = (blockIdx.x * blockDim.x + threadIdx.x) >> 5;

    // A operands: U rows in 16x4 f32 A layout. chunk kc -> K = 4kc+2h, 4kc+2h+1
    v2f aUr[4], aUi[4], aUiN[4];
#pragma unroll
    for (int kc = 0; kc < 4; ++kc) {
        int k0 = m * 16 + 4 * kc + 2 * h;
        v2f tr = {Ur[k0], Ur[k0 + 1]};
        v2f ti = {Ui[k0], Ui[k0 + 1]};
        aUr[kc] = tr;
        aUi[kc] = ti;
        aUiN[kc] = -ti;   // f32 WMMA has no A-negate modifier; pre-negate
    }

    float accS[4] = {0, 0, 0, 0};
    float accQ[4] = {0, 0, 0, 0};

    const int ntiles = Bsz >> 4;
    for (int t = wave; t < ntiles; t += nwaves) {
        const int b = t * 16 + m;
        const float4* f4 = (const float4*)(feat + (size_t)b * 16);
        float4 fa = f4[0], fb = f4[1], fc = f4[2], fd = f4[3];
        float A0[4] = {fa.x, fa.y, fa.z, fa.w};
        float A1[4] = {fb.x, fb.y, fb.z, fb.w};
        float A2[4] = {fc.x, fc.y, fc.z, fc.w};
        float A3[4] = {fd.x, fd.y, fd.z, fd.w};

        float2 V0[2], V1[2], V2[2], V3[2];
        spinor(A0[0], A1[0], A2[0], A3[0], V0[0], V0[1]);
        spinor(A0[1], A1[1], A2[1], A3[1], V1[0], V1[1]);
        spinor(A0[2], A1[2], A2[2], A3[2], V2[0], V2[1]);
        spinor(A0[3], A1[3], A2[3], A3[3], V3[0], V3[1]);

        float2 p01[4], p23[4], enc[16];
#pragma unroll
        for (int i = 0; i < 4; ++i) {
            p01[i] = cmul(V0[i >> 1], V1[i & 1]);
            p23[i] = cmul(V2[i >> 1], V3[i & 1]);
        }
#pragma unroll
        for (int i = 0; i < 16; ++i) enc[i] = cmul(p01[i >> 2], p23[i & 3]);

        // Complex GEMM: Psi = U * Enc  (tile: 16 states x 16 batch)
        v8f cr = {0, 0, 0, 0, 0, 0, 0, 0};
        v8f ci = {0, 0, 0, 0, 0, 0, 0, 0};
#pragma unroll
        for (int kc = 0; kc < 4; ++kc) {
            float2 eA = h ? enc[4 * kc + 2] : enc[4 * kc + 0];
            float2 eB = h ? enc[4 * kc + 3] : enc[4 * kc + 1];
            v2f eR = {eA.x, eB.x};
            v2f eI = {eA.y, eB.y};
            cr = wmma4(aUr[kc],  eR, cr);
            cr = wmma4(aUiN[kc], eI, cr);   // - Ui*Ei
            ci = wmma4(aUr[kc],  eI, ci);
            ci = wmma4(aUi[kc],  eR, ci);
        }

        // <Z_w>: lane holds states 8h..8h+7 of batch m; wire0 = bit3 = h
        float P = 0, e1 = 0, e2 = 0, e3 = 0;
#pragma unroll
        for (int v = 0; v < 8; ++v) {
            float pr = cr[v] * cr[v] + ci[v] * ci[v];
            P += pr;
            e1 += (v & 4) ? -pr : pr;
            e2 += (v & 2) ? -pr : pr;
            e3 += (v & 1) ? -pr : pr;
        }
        float e0 = h ? -P : P;
        e0 += __shfl_xor(e0, 16, 32);
        e1 += __shfl_xor(e1, 16, 32);
        e2 += __shfl_xor(e2, 16, 32);
        e3 += __shfl_xor(e3, 16, 32);

        if (h == 0) {
            ((float4*)ezbuf)[b] = make_float4(e0, e1, e2, e3);
            accS[0] += e0; accS[1] += e1; accS[2] += e2; accS[3] += e3;
            accQ[0] += e0 * e0; accQ[1] += e1 * e1;
            accQ[2] += e2 * e2; accQ[3] += e3 * e3;
        }
    }

    // Deterministic wave reduction (upper half-lanes contribute zeros)
#pragma unroll
    for (int off = 1; off < 32; off <<= 1) {
#pragma unroll
        for (int w = 0; w < 4; ++w) {
            accS[w] += __shfl_xor(accS[w], off, 32);
            accQ[w] += __shfl_xor(accQ[w], off, 32);
        }
    }
    __shared__ float lacc[8][8];
    if (lane == 0) {
        int wv = threadIdx.x >> 5;
        for (int w = 0; w < 4; ++w) {
            lacc[wv][w] = accS[w];
            lacc[wv][4 + w] = accQ[w];
        }
    }
    __syncthreads();
    if (threadIdx.x < 8) {
        float s = 0;
        for (int wv = 0; wv < 8; ++wv) s += lacc[wv][threadIdx.x];
        partial[blockIdx.x * 8 + threadIdx.x] = s;
    }
}

// --------------------------- kernel 3: BN+Linear coefficients --------------
__global__ void stats_kernel(const float* __restrict__ partial, int nblk,
                             const float* __restrict__ gamma,
                             const float* __restrict__ beta,
                             const float* __restrict__ W,
                             const float* __restrict__ bias,
                             float* __restrict__ coef, int Bsz) {
    __shared__ float S[8];
    if (threadIdx.x < 8) {
        float s = 0;
        for (int i = 0; i < nblk; ++i) s += partial[i * 8 + threadIdx.x];
        S[threadIdx.x] = s;
    }
    __syncthreads();
    if (threadIdx.x == 0) {
        float cst = bias[0];
        float invB = 1.0f / (float)Bsz;
        for (int w = 0; w < 4; ++w) {
            float mu = S[w] * invB;
            float var = S[4 + w] * invB - mu * mu;   // biased variance
            float inv = rsqrtf(var + 1e-5f);
            coef[w] = gamma[w] * inv * W[w];
            cst += (beta[w] - mu * inv * gamma[w]) * W[w];
        }
        coef[4] = cst;
    }
}

// --------------------------- kernel 4: output ------------------------------
__global__ void finalize_kernel(const float* __restrict__ ezbuf,
                                const float* __restrict__ coef,
                                float* __restrict__ out, int Bsz) {
    int i = blockIdx.x * blockDim.x + threadIdx.x;
    if (i < Bsz) {
        float4 e = ((const float4*)ezbuf)[i];
        out[i] = e.x * coef[0] + e.y * coef[1] + e.z * coef[2] + e.w * coef[3] + coef[4];
    }
}

// --------------------------- launch ----------------------------------------
extern "C" void kernel_launch(void* const* d_in, const int* in_sizes, int n_in,
                              void* d_out, int out_size, void* d_ws, size_t ws_size,
                              hipStream_t stream) {
    const float* feat  = (const float*)d_in[0];
    const float* rp    = (const float*)d_in[1];
    const float* rx0   = (const float*)d_in[2];
    const float* ry0   = (const float*)d_in[3];
    const float* rz0   = (const float*)d_in[4];
    const float* crx0  = (const float*)d_in[5];
    const float* gamma = (const float*)d_in[6];
    const float* beta  = (const float*)d_in[7];
    const float* W     = (const float*)d_in[8];
    const float* bias  = (const float*)d_in[9];
    float* out = (float*)d_out;

    const int Bsz = in_sizes[0] / 16;

    // Replicate numpy RandomState(42) gate sequence on the host (deterministic,
    // pure CPU; baked into kernel args so graph capture is safe).
    GateOps ops{};
    std::mt19937 mt(42u);  // identical stream to numpy's MT19937 init_genrand
    auto nxt = [&]() -> unsigned { return (unsigned)mt(); };
    // legacy randint(4): masked rejection on 64-bit draw (hi<<32|lo); mask=3 -> lo&3
    auto randint4 = [&]() -> unsigned { (void)nxt(); return nxt() & 3u; };
    // legacy rk_interval(max): 32-bit masked rejection
    auto interval = [&](unsigned mx) -> unsigned {
        unsigned mask = mx;
        mask |= mask >> 1; mask |= mask >> 2; mask |= mask >> 4;
        mask |= mask >> 8; mask |= mask >> 16;
        unsigned v;
        do { v = nxt() & mask; } while (v > mx);
        return v;
    };
    unsigned n = 0, p = 0;
    auto push = [&](unsigned kind, unsigned w0, unsigned w1, unsigned pi) {
        ops.op[n++] = kind | (w0 << 3) | (w1 << 5) | (pi << 7);
    };
    for (int g = 0; g < 50; ++g) {
        unsigned k = randint4();
        if (k < 3) {
            unsigned w = randint4();
            push(k, w, 0, p++);
        } else {
            int a[4] = {0, 1, 2, 3};
            for (int i = 3; i >= 1; --i) {   // Fisher-Yates shuffle of arange(4)
                unsigned j = interval((unsigned)i);
                int tmp = a[i]; a[i] = a[(int)j]; a[(int)j] = tmp;
            }
            push(3, (unsigned)a[0], (unsigned)a[1], 63);
        }
    }
    push(0, 0, 0, 50);  // RX(rx0) wire0
    push(1, 1, 0, 51);  // RY(ry0) wire1
    push(2, 3, 0, 52);  // RZ(rz0) wire3
    push(4, 0, 2, 53);  // CRX(crx0) control0 target2
    push(5, 3, 0, 63);  // H wire3
    push(6, 2, 0, 63);  // SX wire2
    push(3, 3, 0, 63);  // CNOT control3 target0
    ops.n = n;

    // Workspace layout (floats): Ur[256] Ui[256] partial[1024] coef[5] pad, ezbuf[B*4]
    float* Ur = (float*)d_ws;
    float* Ui = Ur + 256;
    float* partial = Ui + 256;
    float* coef = partial + 1024;
    float* ezbuf = (float*)d_ws + 2048;  // 8192-byte offset, float4-aligned

    const int THREADS = 256;
    const int NBLK = 128;
    const int nwaves = NBLK * THREADS / 32;

    build_u_kernel<<<1, 16, 0, stream>>>(ops, rp, rx0, ry0, rz0, crx0, Ur, Ui);
    qsim_main_kernel<<<NBLK, THREADS, 0, stream>>>(feat, Ur, Ui, ezbuf, partial,
                                                   Bsz, nwaves);
    stats_kernel<<<1, 32, 0, stream>>>(partial, NBLK, gamma, beta, W, bias, coef, Bsz);
    finalize_kernel<<<(Bsz + 255) / 256, 256, 0, stream>>>(ezbuf, coef, out, Bsz);
}